// DetectionModel_18631568130280
// MI455X (gfx1250) — compile-verified
//
#include <hip/hip_runtime.h>
#include <cstdint>
#include <cstddef>

// Problem constants (match reference: B=2, N=8192, C=80)
#define BATCH   2
#define NBOX    8192
#define NCLS    80
#define CONF_T  0.5f
#define NMS_T   0.5f

#define SORT_THREADS 1024
#define NMS_THREADS  1024
#define SORT_LDS_BYTES ((size_t)NBOX * 8)                 // 64 KB: packed sort keys
#define NMS_LDS_BYTES  ((size_t)NBOX * 16 + (size_t)NBOX * 3) // 152 KB: float4 boxes + valid/supp/keep bytes

__device__ __forceinline__ float sigf(float x) { return 1.0f / (1.0f + __expf(-x)); }
__device__ __forceinline__ float clamp01(float x) { return fminf(fmaxf(x, 0.0f), 1.0f); }

// ---------------------------------------------------------------------------
// Kernel 1: fused sigmoid / max / argmax / box decode / validity mask.
// One thread per (batch, anchor). float4 (b128) vectorized over the 80 classes.
// ---------------------------------------------------------------------------
__global__ void __launch_bounds__(256)
decode_score_kernel(const float* __restrict__ logits,   // [B,N,C]
                    const float* __restrict__ deltas,   // [B,N,4]
                    const float* __restrict__ anchors,  // [N,4]
                    float* __restrict__ outBoxes,       // [B,N,4]
                    float* __restrict__ outMax,         // [B,N]
                    float* __restrict__ outLab,         // [B,N] (label as float)
                    float* __restrict__ outAll,         // [B,N,C]
                    unsigned* __restrict__ validOut)    // [B,N] ws
{
    const int gid = blockIdx.x * blockDim.x + threadIdx.x;
    if (gid >= BATCH * NBOX) return;
    const int n = gid & (NBOX - 1);

    // CDNA5 prefetch path (global_prefetch_b8) for the row we'll need at the end.
    __builtin_prefetch(deltas + (size_t)gid * 4, 0, 0);
    __builtin_prefetch(anchors + (size_t)n * 4, 0, 0);

    const float4* lg = reinterpret_cast<const float4*>(logits) + (size_t)gid * (NCLS / 4);
    float4*       as = reinterpret_cast<float4*>(outAll)       + (size_t)gid * (NCLS / 4);

    float ml = -3.402823466e38f;
    int   lab = 0;
#pragma unroll
    for (int q = 0; q < NCLS / 4; ++q) {
        float4 v = lg[q];
        const int base = q * 4;
        // strict '>' keeps the FIRST max index, matching jnp.argmax
        if (v.x > ml) { ml = v.x; lab = base + 0; }
        if (v.y > ml) { ml = v.y; lab = base + 1; }
        if (v.z > ml) { ml = v.z; lab = base + 2; }
        if (v.w > ml) { ml = v.w; lab = base + 3; }
        float4 s;
        s.x = sigf(v.x); s.y = sigf(v.y); s.z = sigf(v.z); s.w = sigf(v.w);
        as[q] = s;  // sigmoid is monotonic -> argmax(logits) == argmax(scores)
    }

    const float4 d = reinterpret_cast<const float4*>(deltas)[gid];
    const float4 a = reinterpret_cast<const float4*>(anchors)[n];
    const float aw  = a.z - a.x, ah  = a.w - a.y;
    const float acx = a.x + 0.5f * aw, acy = a.y + 0.5f * ah;
    const float dw  = fminf(d.z, 4.0f), dh = fminf(d.w, 4.0f);
    const float pcx = d.x * aw + acx,   pcy = d.y * ah + acy;
    const float pw  = __expf(dw) * aw,  ph  = __expf(dh) * ah;
    const float x1 = clamp01(pcx - 0.5f * pw);
    const float y1 = clamp01(pcy - 0.5f * ph);
    const float x2 = clamp01(pcx + 0.5f * pw);
    const float y2 = clamp01(pcy + 0.5f * ph);

    float4 bb; bb.x = x1; bb.y = y1; bb.z = x2; bb.w = y2;
    reinterpret_cast<float4*>(outBoxes)[gid] = bb;

    const float ms = sigf(ml);
    outMax[gid] = ms;
    outLab[gid] = (float)lab;

    const float w = x2 - x1, h = y2 - y1;
    const bool valid = (ms > CONF_T) && (w > 0.01f) && (h > 0.01f) && (w < 0.99f) && (h < 0.99f);
    validOut[gid] = valid ? 1u : 0u;
}

// ---------------------------------------------------------------------------
// Kernel 2: per-batch in-LDS bitonic argsort of 8192 scores.
// Key = (~scoreBits << 32) | idx  -> ascending sort == descending score with
// stable (ascending-index) tie break, exactly like jnp.argsort(-scores).
// 64 KB LDS — only viable because CDNA5 gives 320 KB LDS per WGP.
// ---------------------------------------------------------------------------
__global__ void __launch_bounds__(SORT_THREADS)
sort_kernel(const float* __restrict__ maxScores,  // [B,N]
            int* __restrict__ sortedIdx)          // [B,N] ws
{
    extern __shared__ __align__(16) unsigned long long skey[];
    const int b = blockIdx.x, tid = threadIdx.x;

    for (int s = tid; s < NBOX; s += SORT_THREADS) {
        // scores are sigmoid outputs in (0,1): positive -> IEEE bits monotonic
        const unsigned bits = __float_as_uint(maxScores[b * NBOX + s]);
        skey[s] = ((unsigned long long)(~bits) << 32) | (unsigned)s;
    }
    __syncthreads();

    for (unsigned k = 2; k <= NBOX; k <<= 1) {
        for (unsigned j = k >> 1; j > 0; j >>= 1) {
            for (unsigned i = tid; i < NBOX; i += SORT_THREADS) {
                const unsigned ixj = i ^ j;
                if (ixj > i) {
                    const unsigned long long va = skey[i], vb = skey[ixj];
                    const bool up = ((i & k) == 0);
                    if ((va > vb) == up) { skey[i] = vb; skey[ixj] = va; }
                }
            }
            __syncthreads();
        }
    }

    for (int s = tid; s < NBOX; s += SORT_THREADS)
        sortedIdx[b * NBOX + s] = (int)(unsigned)(skey[s] & 0xFFFFFFFFull);
}

// ---------------------------------------------------------------------------
// Kernel 3: exact greedy NMS, one workgroup per batch, everything in LDS.
// Boxes are GATHERED in sorted order straight into LDS with CDNA5 async
// global->LDS b128 loads (ASYNCcnt), then the sequential keep scan runs with
// 1024-thread-parallel suppression. 152 KB dynamic LDS.
// ---------------------------------------------------------------------------
__global__ void __launch_bounds__(NMS_THREADS)
nms_kernel(const float* __restrict__ boxes,      // [B,N,4] (already decoded+clipped)
           const unsigned* __restrict__ valid,   // [B,N]
           const int* __restrict__ sortedIdx,    // [B,N]
           float* __restrict__ keepOut)          // [B,N] (bool as 0/1 float)
{
    extern __shared__ __align__(16) char smemRaw[];
    float4*        boxS   = reinterpret_cast<float4*>(smemRaw);         // 128 KB
    unsigned char* validS = reinterpret_cast<unsigned char*>(smemRaw + (size_t)NBOX * 16);
    unsigned char* suppS  = validS + NBOX;
    unsigned char* keepS  = suppS + NBOX;
    __shared__ int sPivot;

    const int b = blockIdx.x, tid = threadIdx.x;
    const size_t rowOff = (size_t)b * NBOX;

    // Async-gather sorted boxes into LDS (per-lane global addr -> per-lane LDS addr).
    for (int s = tid; s < NBOX; s += NMS_THREADS) {
        const int orig = sortedIdx[rowOff + s];
        const float* g = boxes + (rowOff + (size_t)orig) * 4;
        const unsigned ldsAddr = (unsigned)(uintptr_t)(&boxS[s]);
        asm volatile("global_load_async_to_lds_b128 %0, %1, off"
                     :: "v"(ldsAddr), "v"(g)
                     : "memory");
        validS[s] = (unsigned char)valid[rowOff + orig];
        suppS[s]  = 0;
    }
    asm volatile("s_wait_asynccnt 0" ::: "memory");
    __syncthreads();

    // Greedy scan: keep[i] = valid[i] && no previously-KEPT box overlaps > thr.
    int i = 0;
    while (i < NBOX) {
        if (tid == 0) {
            int p = i;
            while (p < NBOX && (!validS[p] || suppS[p])) { keepS[p] = 0; ++p; }
            if (p < NBOX) keepS[p] = 1;
            sPivot = p;
        }
        __syncthreads();
        const int p = sPivot;
        if (p >= NBOX) break;

        const float4 pb = boxS[p];
        const float pArea = (pb.z - pb.x) * (pb.w - pb.y);
        for (int j = p + 1 + tid; j < NBOX; j += NMS_THREADS) {
            if (suppS[j]) continue;
            const float4 q = boxS[j];
            const float ix1 = fmaxf(pb.x, q.x), iy1 = fmaxf(pb.y, q.y);
            const float ix2 = fminf(pb.z, q.z), iy2 = fminf(pb.w, q.w);
            const float inter = fmaxf(ix2 - ix1, 0.0f) * fmaxf(iy2 - iy1, 0.0f);
            const float uni = pArea + (q.z - q.x) * (q.w - q.y) - inter;
            // iou > thr  <=>  inter > thr * max(uni, 1e-9)   (division-free)
            if (inter > NMS_T * fmaxf(uni, 1e-9f)) suppS[j] = 1;
        }
        __syncthreads();
        i = p + 1;
    }
    __syncthreads();

    // Scatter keep flags back to original anchor order.
    for (int s = tid; s < NBOX; s += NMS_THREADS) {
        const int orig = sortedIdx[rowOff + s];
        keepOut[rowOff + orig] = keepS[s] ? 1.0f : 0.0f;
    }
}

// ---------------------------------------------------------------------------
// Host launcher. Output layout (flat, reference return order):
//   boxes[B*N*4] | max_scores[B*N] | labels[B*N] | keep[B*N] | all_scores[B*N*C]
// Workspace: sortedIdx int[B*N] (64 KB) | valid u32[B*N] (64 KB)
// ---------------------------------------------------------------------------
extern "C" void kernel_launch(void* const* d_in, const int* in_sizes, int n_in,
                              void* d_out, int out_size, void* d_ws, size_t ws_size,
                              hipStream_t stream)
{
    (void)in_sizes; (void)n_in; (void)out_size; (void)ws_size;

    const float* logits  = (const float*)d_in[0];
    const float* deltas  = (const float*)d_in[1];
    const float* anchors = (const float*)d_in[2];

    float* outBoxes = (float*)d_out;
    float* outMax   = outBoxes + (size_t)BATCH * NBOX * 4;
    float* outLab   = outMax   + (size_t)BATCH * NBOX;
    float* outKeep  = outLab   + (size_t)BATCH * NBOX;
    float* outAll   = outKeep  + (size_t)BATCH * NBOX;

    int*      sortedIdx = (int*)d_ws;
    unsigned* validWs   = (unsigned*)((char*)d_ws + (size_t)BATCH * NBOX * sizeof(int));

    // Raise dynamic-LDS caps (>64 KB). Idempotent & graph-capture safe.
    (void)hipFuncSetAttribute(reinterpret_cast<const void*>(sort_kernel),
                              hipFuncAttributeMaxDynamicSharedMemorySize, (int)SORT_LDS_BYTES);
    (void)hipFuncSetAttribute(reinterpret_cast<const void*>(nms_kernel),
                              hipFuncAttributeMaxDynamicSharedMemorySize, (int)NMS_LDS_BYTES);

    const int threads1 = 256;
    const int blocks1  = (BATCH * NBOX + threads1 - 1) / threads1;
    decode_score_kernel<<<blocks1, threads1, 0, stream>>>(
        logits, deltas, anchors, outBoxes, outMax, outLab, outAll, validWs);

    sort_kernel<<<BATCH, SORT_THREADS, SORT_LDS_BYTES, stream>>>(outMax, sortedIdx);

    nms_kernel<<<BATCH, NMS_THREADS, NMS_LDS_BYTES, stream>>>(
        outBoxes, validWs, sortedIdx, outKeep);
}